// ChamferDistance_18949395710666
// MI455X (gfx1250) — compile-verified
//
#include <hip/hip_runtime.h>
#include <float.h>

typedef __attribute__((ext_vector_type(2))) float v2f;
typedef __attribute__((ext_vector_type(8))) float v8f;

// Pack (x, y, z, ||p||^2) into float4 for single-b64-per-lane B-operand loads.
__global__ void chamfer_pack_kernel(const float* __restrict__ xyz,
                                    float4* __restrict__ packed, int total) {
  int i = blockIdx.x * blockDim.x + threadIdx.x;
  if (i < total) {
    float x = xyz[3 * i + 0];
    float y = xyz[3 * i + 1];
    float z = xyz[3 * i + 2];
    packed[i] = make_float4(x, y, z, x * x + y * y + z * z);
  }
}

// One wave32 handles a 16-row query tile; streams all nP points in 16-col tiles.
// D = A*B + C with A=(-2x1,-2y1,-2z1,1), B=(x2,y2,z2,sq2), C=sq1 -> D = squared dist.
template <bool WITH_IDX>
__global__ __launch_bounds__(256) void chamfer_nn_kernel(
    const float4* __restrict__ packedQ, const float4* __restrict__ packedP,
    float* __restrict__ dist_out, int* __restrict__ idx_out, int nQ, int nP) {
  const int lane = threadIdx.x & 31;
  const int wave = threadIdx.x >> 5;
  const int gwave = blockIdx.x * (blockDim.x >> 5) + wave;
  const int tilesPerBatch = nQ >> 4;
  const int batch = gwave / tilesPerBatch;
  const int tile = gwave % tilesPerBatch;
  const int qbase = batch * nQ + tile * 16;  // flat index of first query row

  const bool hi = lane >= 16;
  const int n_lane = lane & 15;
  const int half8 = hi ? 8 : 0;

  // A operand: lanes 0-15 -> (K0,K1)=(-2x,-2y) of row lane; lanes 16-31 -> (K2,K3)=(-2z, 1).
  float4 q = packedQ[qbase + n_lane];
  v2f a;
  a.x = hi ? (-2.0f * q.z) : (-2.0f * q.x);
  a.y = hi ? 1.0f : (-2.0f * q.y);

  // C operand: VGPR r holds row (r + 8*hi) -> sq1 of that row, pulled via intra-half shuffle.
  v8f c;
#pragma unroll
  for (int r = 0; r < 8; ++r) c[r] = __shfl(q.w, r + half8, 16);

  // B operand pointer: lane n<16 reads (x,y) of col n; lane>=16 reads (z,sq) of col n.
  const float* pB = reinterpret_cast<const float*>(packedP + (size_t)batch * nP) +
                    (size_t)n_lane * 4 + (hi ? 2 : 0);

  float run[8];
  int tmin[8];
#pragma unroll
  for (int r = 0; r < 8; ++r) {
    run[r] = FLT_MAX;
    tmin[r] = 0;
  }

  const int ntiles = nP >> 4;
#pragma unroll 4
  for (int t = 0; t < ntiles; ++t) {
    v2f b = *reinterpret_cast<const v2f*>(pB);  // global_load_b64
    pB += 64;                                   // 16 points * 4 floats
    // D = A x B + C : full squared-distance 16x16 tile in 8 VGPRs.
    v8f d = __builtin_amdgcn_wmma_f32_16x16x4_f32(false, a, false, b, (short)0, c,
                                                  false, false);
#pragma unroll
    for (int r = 0; r < 8; ++r) {
      if (WITH_IDX) {
        bool lt = d[r] < run[r];
        run[r] = lt ? d[r] : run[r];
        tmin[r] = lt ? t : tmin[r];
      } else {
        run[r] = fminf(run[r], d[r]);
      }
    }
  }

  // Cross-lane argmin over the 16 lanes of each half (xor masks stay inside halves).
#pragma unroll
  for (int r = 0; r < 8; ++r) {
    float dmin = run[r];
    int imin = WITH_IDX ? (tmin[r] * 16 + n_lane) : 0;
#pragma unroll
    for (int mask = 1; mask < 16; mask <<= 1) {
      float od = __shfl_xor(dmin, mask, 32);
      int oi = __shfl_xor(imin, mask, 32);
      bool take = (od < dmin) || (WITH_IDX && (od == dmin) && (oi < imin));
      dmin = take ? od : dmin;
      imin = take ? oi : imin;
    }
    if (n_lane == 0) {
      int row = qbase + r + half8;
      dist_out[row] = dmin;
      if (WITH_IDX) idx_out[row] = imin;
    }
  }
}

extern "C" void kernel_launch(void* const* d_in, const int* in_sizes, int n_in,
                              void* d_out, int out_size, void* d_ws, size_t ws_size,
                              hipStream_t stream) {
  const float* xyz1 = (const float*)d_in[0];
  const float* xyz2 = (const float*)d_in[1];
  const int N = 8192, M = 8192;
  const int B = in_sizes[0] / (3 * N);  // = 4

  float4* packed1 = (float4*)d_ws;
  float4* packed2 = packed1 + (size_t)B * N;
  const int totalQ = B * N;
  const int totalP = B * M;

  chamfer_pack_kernel<<<(totalQ + 255) / 256, 256, 0, stream>>>(xyz1, packed1, totalQ);
  chamfer_pack_kernel<<<(totalP + 255) / 256, 256, 0, stream>>>(xyz2, packed2, totalP);

  // Output layout: dist1 [B*N] | dist2 [B*M] | idx1 [B*N] (int32 bits).
  float* dist1 = (float*)d_out;
  float* dist2 = dist1 + totalQ;
  int* idx1 = (int*)(dist2 + totalP);

  const int wavesPerBlock = 256 / 32;  // 8 waves, 16 rows each -> 128 rows/block
  const int blocks1 = (B * N / 16) / wavesPerBlock;
  const int blocks2 = (B * M / 16) / wavesPerBlock;

  chamfer_nn_kernel<true><<<blocks1, 256, 0, stream>>>(packed1, packed2, dist1, idx1, N, M);
  chamfer_nn_kernel<false><<<blocks2, 256, 0, stream>>>(packed2, packed1, dist2, nullptr, M, N);
}